// ExponentialFamilyParticleFilter_42666205118528
// MI455X (gfx1250) — compile-verified
//
#include <hip/hip_runtime.h>
#include <hip/hip_bf16.h>
#include <math.h>

#define BB 16
#define TT 1024
#define DD 256
#define KK 64

typedef float v2f __attribute__((ext_vector_type(2)));
typedef float v8f __attribute__((ext_vector_type(8)));

__global__ void init_out_kernel(float* out) { out[0] = 0.0f; }

// Per-lane async copy: 4 bytes global -> this lane's private LDS slot.
// GV addressing: 64-bit per-lane address in a VGPR pair, saddr = off.
// Tracked by ASYNCcnt (gfx1250 async path), no VGPR return traffic.
__device__ __forceinline__ void async_row_to_lds(const float* row_base, float* lds_slot) {
  uint32_t ldsaddr = (uint32_t)(uintptr_t)lds_slot;                    // LDS byte address
  uint64_t gaddr   = (uint64_t)(uintptr_t)(row_base + threadIdx.x);   // per-lane global addr
  asm volatile("global_load_async_to_lds_b32 %0, %1, off"
               :: "v"(ldsaddr), "v"(gaddr)
               : "memory");
}
__device__ __forceinline__ void wait_async0() {
  asm volatile("s_wait_asynccnt 0x0" ::: "memory");
}

__global__ __launch_bounds__(256) void crp_nig_kernel(
    const float* __restrict__ X, const int* __restrict__ z,
    const float* __restrict__ loc, const float* __restrict__ log_conc,
    const float* __restrict__ log_scale, const float* __restrict__ spl,
    float* __restrict__ out)
{
  __shared__ int   s_z[TT];
  __shared__ int   s_list[TT];
  __shared__ int   s_scan[DD];
  __shared__ float s_x[2][DD];
  __shared__ float s_red[DD];

  const int tid = threadIdx.x;
  const int b   = blockIdx.x >> 6;   // blockIdx = b*64 + k
  const int k   = blockIdx.x & 63;

  // ---- load z row (1024 ints) : one b128 per thread ----
  {
    const int4* z4 = (const int4*)(z + b * TT);
    ((int4*)s_z)[tid] = z4[tid];
  }
  __syncthreads();

  // ---- stream-compact the timesteps where z[b,t]==k (order preserved) ----
  int cnt = 0; int myt[4];
  #pragma unroll
  for (int j = 0; j < 4; ++j) {
    int t = tid * 4 + j;
    if (s_z[t] == k) myt[cnt++] = t;
  }
  s_scan[tid] = cnt;
  __syncthreads();
  for (int off = 1; off < DD; off <<= 1) {          // Hillis-Steele inclusive scan
    int v = (tid >= off) ? s_scan[tid - off] : 0;
    __syncthreads();
    s_scan[tid] += v;
    __syncthreads();
  }
  const int excl  = s_scan[tid] - cnt;
  const int total = s_scan[DD - 1];

  // CRP term is closed-form per occurrence index g at time t (ALPHA = 1):
  //   crp = log(max(g,1)) - log(t+1)   -> computed fully in parallel here.
  float crp_acc = 0.0f;
  for (int j = 0; j < cnt; ++j) {
    int g = excl + j;
    int t = myt[j];
    s_list[g] = t;
    crp_acc += logf(fmaxf((float)g, 1.0f)) - logf((float)t + 1.0f);
  }
  __syncthreads();

  // ---- per-dim NIG/Beta state lives in registers (thread d = tid) ----
  float conc = expf(log_conc[tid]);
  float m    = loc[tid];
  float kap  = 2.0f * conc + 3.0f;
  float a    = conc;
  float bv   = expf(log_scale[tid]);
  float c1   = kap + 1.0f;
  float c0   = (kap + 1.0f) * expf(spl[tid]);

  const float* Xb = X + (size_t)b * TT * DD;
  float acc = crp_acc;

  // ---- serial chain over this cluster's occurrences, async double-buffered ----
  if (total > 0) async_row_to_lds(Xb + (size_t)s_list[0] * DD, &s_x[0][tid]);

  for (int i = 0; i < total; ++i) {
    wait_async0();                                  // row i landed in LDS
    float x = s_x[i & 1][tid];
    if (i + 1 < total)                              // prefetch row i+1 behind the math
      async_row_to_lds(Xb + (size_t)s_list[i + 1] * DD, &s_x[(i + 1) & 1][tid]);

    bool  y  = x > 0.0f;
    float tl = y ? logf(x) : 0.0f;
    float s  = (y ? logf(c1) : logf(c0)) - logf(c1 + c0);   // Beta-Bernoulli predictive
    if (y) {
      float kap1 = kap + 1.0f;
      float sig2 = bv * kap1 / (a * kap);
      float d0   = tl - m;
      float nu   = 2.0f * a;
      // Student-t log pdf on log x, plus -log x Jacobian
      float lt = lgammaf(a + 0.5f) - lgammaf(a)
               - 0.5f * logf(nu * 3.14159265358979323846f)
               - 0.5f * logf(sig2)
               - (a + 0.5f) * log1pf((d0 * d0 / sig2) / nu);
      s += lt - tl;
      // conjugate updates (use pre-update kap, m)
      bv += kap * d0 * d0 / (2.0f * kap1);
      m   = (kap * m + tl) / kap1;
      kap = kap1;
      a  += 0.5f;
      c1 += 1.0f;
    } else {
      c0 += 1.0f;
    }
    acc += s;
  }

  // ---- block reduction: LDS tree to 64 partials, then WMMA ones-matmul ----
  s_red[tid] = acc;
  __syncthreads();
  if (tid < 128) s_red[tid] += s_red[tid + 128];
  __syncthreads();
  if (tid < 64)  s_red[tid] += s_red[tid + 64];
  __syncthreads();

  if (tid < 32) {   // wave 0, full EXEC (wave32) as WMMA requires
    // B (4x16 f32): B[r][n] = s_red[16*r + n]; ones-A makes D[i][n] = column sums
    int base0 = (tid & 15) + ((tid & 16) ? 32 : 0);
    v2f bmat; bmat[0] = s_red[base0]; bmat[1] = s_red[base0 + 16];
    v2f amat; amat[0] = 1.0f; amat[1] = 1.0f;
    v8f cmat = {};
    v8f dmat = __builtin_amdgcn_wmma_f32_16x16x4_f32(
        /*neg_a=*/false, amat, /*neg_b=*/false, bmat,
        /*c_mod=*/(short)0, cmat, /*reuse_a=*/false, /*reuse_b=*/false);
    float v = dmat[0];                 // lane n holds column-sum n (rows identical)
    v += __shfl_xor(v, 1, 32);
    v += __shfl_xor(v, 2, 32);
    v += __shfl_xor(v, 4, 32);
    v += __shfl_xor(v, 8, 32);         // lane 0: sum of the 16 column sums
    if (tid == 0) atomicAdd(out, -v * (1.0f / (float)BB));
  }
}

extern "C" void kernel_launch(void* const* d_in, const int* in_sizes, int n_in,
                              void* d_out, int out_size, void* d_ws, size_t ws_size,
                              hipStream_t stream) {
  (void)in_sizes; (void)n_in; (void)out_size; (void)d_ws; (void)ws_size;
  const float* X      = (const float*)d_in[0];
  const int*   z      = (const int*)  d_in[1];
  const float* loc    = (const float*)d_in[2];
  const float* lconc  = (const float*)d_in[3];
  const float* lscale = (const float*)d_in[4];
  const float* spl    = (const float*)d_in[5];
  float* out = (float*)d_out;

  init_out_kernel<<<dim3(1), dim3(1), 0, stream>>>(out);
  crp_nig_kernel<<<dim3(BB * KK), dim3(DD), 0, stream>>>(
      X, z, loc, lconc, lscale, spl, out);
}